// StableGATLayer_26482768347669
// MI455X (gfx1250) — compile-verified
//
#include <hip/hip_runtime.h>
#include <math.h>

#define HEADS 8
#define DIM 128
#define EPS_LAP 1e-6f
#define EPS_LN 1e-5f
#define NEG_SLOPE 0.2f

typedef float v2f __attribute__((ext_vector_type(2)));
typedef float v8f __attribute__((ext_vector_type(8)));

// ---------------------------------------------------------------------------
// float atomic-max via signed-int max / unsigned-int min trick.
// Init bit pattern 0xFFFFFFFF: as signed int it is -1 (any l>=0 replaces);
// as unsigned it is UINT_MAX (any negative-float pattern replaces via min).
// Every dst gets at least one real update (self-loop), so init never survives.
// ---------------------------------------------------------------------------
__device__ __forceinline__ void atomicMaxFloat(float* addr, float v) {
  if (v >= 0.0f) atomicMax((int*)addr, __float_as_int(v));
  else           atomicMin((unsigned int*)addr, __float_as_uint(v));
}

__device__ __forceinline__ float waveAllReduceAdd(float s) {
  #pragma unroll
  for (int o = 16; o > 0; o >>= 1) s += __shfl_xor(s, o, 32);
  return s;
}

// ---------------------------------------------------------------------------
// WMMA f32 GEMM: Out[16 x 16-tile] = A(row-major, lda=KTOT) x B(row-major, ldb=ldo)
// One wave per 16x16 output tile; 8 waves/block share a 16xKTOT A-tile in LDS.
// V_WMMA_F32_16X16X4_F32 VGPR mapping (ISA 7.12.2):
//   A 16x4: lane<16 -> {K0,K1}, lane>=16 -> {K2,K3}, M = lane&15
//   B 4x16: mirrored (K rows striped across lane halves), N = lane&15
//   C/D   : VGPR r -> M = r + 8*(lane>>4), N = lane&15
// EPI: 0 = plain store, 1 = exact GELU(c + bias), 2 = c + bias + residual
// ---------------------------------------------------------------------------
template<int KTOT, int EPI>
__global__ __launch_bounds__(256)
void wmma_gemm(const float* __restrict__ A, const float* __restrict__ B,
               const float* __restrict__ bias, const float* __restrict__ res,
               float* __restrict__ Out, int ldo) {
  __shared__ float lds[16 * (KTOT + 1)];   // +1 pad: no LDS bank conflicts
  const int tid  = threadIdx.x;
  const int wave = tid >> 5;
  const int lane = tid & 31;
  const int row0 = blockIdx.x * 16;
  const int col0 = (blockIdx.y * 8 + wave) * 16;

  // coalesced stage of the 16 x KTOT A tile
  for (int idx = tid; idx < 16 * KTOT; idx += 256) {
    int r = idx / KTOT, c = idx - r * KTOT;
    lds[r * (KTOT + 1) + c] = A[(size_t)(row0 + r) * KTOT + c];
  }
  __syncthreads();

  const int n  = lane & 15;
  const int m  = lane & 15;
  const int kk = (lane >> 4) * 2;
  v8f acc = {0.f, 0.f, 0.f, 0.f, 0.f, 0.f, 0.f, 0.f};
  #pragma unroll 8
  for (int k = 0; k < KTOT; k += 4) {
    v2f a, b;
    a.x = lds[m * (KTOT + 1) + k + kk];
    a.y = lds[m * (KTOT + 1) + k + kk + 1];
    b.x = B[(size_t)(k + kk) * ldo + col0 + n];
    b.y = B[(size_t)(k + kk + 1) * ldo + col0 + n];
    // (neg_a, A, neg_b, B, c_mod, C, reuse_a, reuse_b)
    acc = __builtin_amdgcn_wmma_f32_16x16x4_f32(false, a, false, b,
                                                (short)0, acc, false, false);
  }

  const int mbase = (lane >> 4) * 8;
  #pragma unroll
  for (int r = 0; r < 8; ++r) {
    int row = row0 + mbase + r;
    int col = col0 + n;
    float v = acc[r];
    if (EPI == 1) {                       // bias + exact GELU
      v += bias[col];
      v = 0.5f * v * (1.0f + erff(v * 0.70710678118654752440f));
    } else if (EPI == 2) {                // bias + residual
      v += bias[col] + res[(size_t)row * DIM + col];
    }
    Out[(size_t)row * ldo + col] = v;
  }
}

// a_s[n,h] = <h[n, h*16:h*16+16], att_src[h]> ; same for a_d
__global__ void attn_coeffs(const float* __restrict__ h,
                            const float* __restrict__ att_src,
                            const float* __restrict__ att_dst,
                            float* __restrict__ a_s, float* __restrict__ a_d, int N) {
  int idx = blockIdx.x * blockDim.x + threadIdx.x;
  if (idx >= N * HEADS) return;
  int hh = idx & (HEADS - 1);
  int i  = idx >> 3;
  const float* hp = h + (size_t)i * DIM + hh * 16;
  float ss = 0.f, dd = 0.f;
  #pragma unroll
  for (int c = 0; c < 16; ++c) {
    ss += hp[c] * att_src[hh * 16 + c];
    dd += hp[c] * att_dst[hh * 16 + c];
  }
  a_s[idx] = ss;
  a_d[idx] = dd;
}

// Pass 1 over (edges + self loops): per-dst segment max of leaky-relu logits.
// Max includes ALL original edges (even src==dst, like the reference). Also
// accumulates out-degree deg[src] over original edges for the Laplacian norm.
__global__ void edge_max_deg(const int* __restrict__ src, const int* __restrict__ dst,
                             const float* __restrict__ a_s, const float* __restrict__ a_d,
                             float* __restrict__ m, float* __restrict__ deg, int E, int N) {
  int tot = E + N;
  for (int e = blockIdx.x * blockDim.x + threadIdx.x; e < tot;
       e += gridDim.x * blockDim.x) {
    int s, d;
    if (e < E) { s = src[e]; d = dst[e]; atomicAdd(&deg[s], 1.0f); }
    else       { s = d = e - E; }
    #pragma unroll
    for (int hh = 0; hh < HEADS; ++hh) {
      float l = a_s[s * HEADS + hh] + a_d[d * HEADS + hh];
      l = l > 0.f ? l : NEG_SLOPE * l;
      atomicMaxFloat(&m[d * HEADS + hh], l);
    }
  }
}

// Pass 2: softmax denominator. Original self-loop edges dropped (keep mask),
// the appended per-node self loop always counted.
__global__ void edge_expsum(const int* __restrict__ src, const int* __restrict__ dst,
                            const float* __restrict__ a_s, const float* __restrict__ a_d,
                            const float* __restrict__ m, float* __restrict__ den,
                            int E, int N) {
  int tot = E + N;
  for (int e = blockIdx.x * blockDim.x + threadIdx.x; e < tot;
       e += gridDim.x * blockDim.x) {
    int s, d;
    if (e < E) { s = src[e]; d = dst[e]; if (s == d) continue; }
    else       { s = d = e - E; }
    #pragma unroll
    for (int hh = 0; hh < HEADS; ++hh) {
      float l = a_s[s * HEADS + hh] + a_d[d * HEADS + hh];
      l = l > 0.f ? l : NEG_SLOPE * l;
      atomicAdd(&den[d * HEADS + hh], expf(l - m[d * HEADS + hh]));
    }
  }
}

// Pass 3 (the only full-width edge pass): gat[d] += coef * h[s], 128 lanes/edge.
__global__ __launch_bounds__(256)
void edge_aggregate(const int* __restrict__ src, const int* __restrict__ dst,
                    const float* __restrict__ a_s, const float* __restrict__ a_d,
                    const float* __restrict__ m, const float* __restrict__ den,
                    const float* __restrict__ h, float* __restrict__ gat, int E) {
  int c   = threadIdx.x & 127;
  int hh  = c >> 4;
  int grp = (blockIdx.x * blockDim.x + threadIdx.x) >> 7;
  int ngr = (gridDim.x * blockDim.x) >> 7;
  for (int e = grp; e < E; e += ngr) {
    int s = src[e], d = dst[e];
    if (s == d) continue;                 // uniform across the 128-lane group
    float l = a_s[s * HEADS + hh] + a_d[d * HEADS + hh];
    l = l > 0.f ? l : NEG_SLOPE * l;
    float coef = expf(l - m[d * HEADS + hh]) / (den[d * HEADS + hh] + 1e-16f);
    atomicAdd(&gat[(size_t)d * DIM + c], coef * h[(size_t)s * DIM + c]);
  }
}

// Self-loop contribution (no atomics needed: kernels serialize on stream) + bias
__global__ void self_loop_bias(const float* __restrict__ a_s, const float* __restrict__ a_d,
                               const float* __restrict__ m, const float* __restrict__ den,
                               const float* __restrict__ h, const float* __restrict__ gat_bias,
                               float* __restrict__ gat, int N) {
  int idx = blockIdx.x * blockDim.x + threadIdx.x;
  if (idx >= N * DIM) return;
  int c = idx & 127, i = idx >> 7, hh = c >> 4;
  float l = a_s[i * HEADS + hh] + a_d[i * HEADS + hh];
  l = l > 0.f ? l : NEG_SLOPE * l;
  float coef = expf(l - m[i * HEADS + hh]) / (den[i * HEADS + hh] + 1e-16f);
  gat[idx] += coef * h[idx] + gat_bias[c];
}

// wave-per-row: r[i] = sum_c gat[i][c]  (lets the mean SpMM be scalar)
__global__ void row_sum(const float* __restrict__ gat, float* __restrict__ r, int N) {
  int row  = (blockIdx.x * blockDim.x + threadIdx.x) >> 5;
  int lane = threadIdx.x & 31;
  if (row >= N) return;
  float s = 0.f;
  #pragma unroll
  for (int j = 0; j < 4; ++j) s += gat[(size_t)row * DIM + lane + 32 * j];
  s = waveAllReduceAdd(s);
  if (lane == 0) r[row] = s;
}

// scalar SpMM over ORIGINAL edges: acc[src] += w_{sd} * val[dst]
__global__ void spmm_scalar(const int* __restrict__ src, const int* __restrict__ dst,
                            const float* __restrict__ deg, const float* __restrict__ val,
                            float* __restrict__ acc, int E) {
  for (int e = blockIdx.x * blockDim.x + threadIdx.x; e < E;
       e += gridDim.x * blockDim.x) {
    int s = src[e], d = dst[e];
    float w = rsqrtf(deg[s] + EPS_LAP) * rsqrtf(deg[d] + EPS_LAP);
    atomicAdd(&acc[s], w * val[d]);
  }
}

// mean[i] = macc[i]/128 ; q[i] = sum_c (gat[i][c]-mean[i])^2  (for the var SpMM)
__global__ void row_q(const float* __restrict__ gat, const float* __restrict__ macc,
                      float* __restrict__ mean, float* __restrict__ q, int N) {
  int row  = (blockIdx.x * blockDim.x + threadIdx.x) >> 5;
  int lane = threadIdx.x & 31;
  if (row >= N) return;
  float mu = macc[row] * (1.0f / DIM);
  float s = 0.f;
  #pragma unroll
  for (int j = 0; j < 4; ++j) {
    float t = gat[(size_t)row * DIM + lane + 32 * j] - mu;
    s += t * t;
  }
  s = waveAllReduceAdd(s);
  if (lane == 0) { mean[row] = mu; q[row] = s; }
}

// Fused: Laplacian-norm affine + residual + LayerNorm (wave per row)
__global__ void lap_ln(const float* __restrict__ gat, const float* __restrict__ mean,
                       const float* __restrict__ vacc,
                       const float* __restrict__ ln1s, const float* __restrict__ ln1b,
                       const float* __restrict__ ln2s, const float* __restrict__ ln2b,
                       const float* __restrict__ x, float* __restrict__ out2, int N) {
  int row  = (blockIdx.x * blockDim.x + threadIdx.x) >> 5;
  int lane = threadIdx.x & 31;
  if (row >= N) return;
  float mu  = mean[row];
  float var = vacc[row] * (1.0f / DIM);
  float inv = 1.0f / sqrtf(var + EPS_LAP);
  float t[4], s = 0.f;
  #pragma unroll
  for (int j = 0; j < 4; ++j) {
    int c = lane + 32 * j;
    float xn = (gat[(size_t)row * DIM + c] - mu) * inv;
    t[j] = xn * ln1s[c] + ln1b[c] + x[(size_t)row * DIM + c];
    s += t[j];
  }
  float mu2 = waveAllReduceAdd(s) * (1.0f / DIM);
  float vs = 0.f;
  #pragma unroll
  for (int j = 0; j < 4; ++j) { float d = t[j] - mu2; vs += d * d; }
  float var2 = waveAllReduceAdd(vs) * (1.0f / DIM);
  float inv2 = rsqrtf(var2 + EPS_LN);
  #pragma unroll
  for (int j = 0; j < 4; ++j) {
    int c = lane + 32 * j;
    out2[(size_t)row * DIM + c] = (t[j] - mu2) * inv2 * ln2s[c] + ln2b[c];
  }
}

// Final row L2 normalize, in place on d_out (wave per row)
__global__ void l2norm(float* __restrict__ out, int N) {
  int row  = (blockIdx.x * blockDim.x + threadIdx.x) >> 5;
  int lane = threadIdx.x & 31;
  if (row >= N) return;
  float s = 0.f;
  float v[4];
  #pragma unroll
  for (int j = 0; j < 4; ++j) {
    v[j] = out[(size_t)row * DIM + lane + 32 * j];
    s += v[j] * v[j];
  }
  s = waveAllReduceAdd(s);
  float invn = 1.0f / fmaxf(sqrtf(s), 1e-12f);
  #pragma unroll
  for (int j = 0; j < 4; ++j)
    out[(size_t)row * DIM + lane + 32 * j] = v[j] * invn;
}

// ---------------------------------------------------------------------------
extern "C" void kernel_launch(void* const* d_in, const int* in_sizes, int n_in,
                              void* d_out, int out_size, void* d_ws, size_t ws_size,
                              hipStream_t stream) {
  const float* x        = (const float*)d_in[0];
  const int*   ei       = (const int*)  d_in[1];   // harness: integer -> const int*
  const float* W        = (const float*)d_in[2];
  const float* att_src  = (const float*)d_in[3];
  const float* att_dst  = (const float*)d_in[4];
  const float* gat_bias = (const float*)d_in[5];
  const float* ln1s     = (const float*)d_in[6];
  const float* ln1b     = (const float*)d_in[7];
  const float* ln2s     = (const float*)d_in[8];
  const float* ln2b     = (const float*)d_in[9];
  const float* W1       = (const float*)d_in[10];
  const float* b1       = (const float*)d_in[11];
  const float* W2       = (const float*)d_in[12];
  const float* b2       = (const float*)d_in[13];

  const int N = in_sizes[0] / DIM;          // 50000 (divisible by 16)
  const int E = in_sizes[1] / 2;            // 800000
  const int* src = ei;
  const int* dst = ei + E;

  // workspace layout (~136 MB of f32)
  float* p      = (float*)d_ws;
  float* h      = p; p += (size_t)N * DIM;
  float* gat    = p; p += (size_t)N * DIM;
  float* out2   = p; p += (size_t)N * DIM;
  float* hidden = p; p += (size_t)N * 2 * DIM;
  float* a_s    = p; p += (size_t)N * HEADS;
  float* a_d    = p; p += (size_t)N * HEADS;
  float* m      = p; p += (size_t)N * HEADS;
  float* den    = p; p += (size_t)N * HEADS;
  float* deg    = p; p += N;   // deg, macc, vacc contiguous for one memset
  float* macc   = p; p += N;
  float* vacc   = p; p += N;
  float* meanv  = p; p += N;
  float* rsum   = p; p += N;
  float* qv     = p; p += N;

  // accumulator init (graph-capturable)
  hipMemsetAsync(gat, 0,    (size_t)N * DIM * sizeof(float), stream);
  hipMemsetAsync(den, 0,    (size_t)N * HEADS * sizeof(float), stream);
  hipMemsetAsync(deg, 0,    (size_t)3 * N * sizeof(float), stream);
  hipMemsetAsync(m,   0xFF, (size_t)N * HEADS * sizeof(float), stream);

  const dim3 blk(256);
  const int rowTiles = N / 16;
  const int waveBlocks = (N * 32 + 255) / 256;

  // 1) h = x @ W  (WMMA f32, K=128, cols=128 -> 8 waves/block)
  wmma_gemm<DIM, 0><<<dim3(rowTiles, 1), blk, 0, stream>>>(x, W, nullptr, nullptr, h, DIM);
  // 2) per-head attention scalars
  attn_coeffs<<<(N * HEADS + 255) / 256, blk, 0, stream>>>(h, att_src, att_dst, a_s, a_d, N);
  // 3) edge softmax (max, sum, aggregate) + self loops
  edge_max_deg <<<2048, blk, 0, stream>>>(src, dst, a_s, a_d, m, deg, E, N);
  edge_expsum  <<<2048, blk, 0, stream>>>(src, dst, a_s, a_d, m, den, E, N);
  edge_aggregate<<<4096, blk, 0, stream>>>(src, dst, a_s, a_d, m, den, h, gat, E);
  self_loop_bias<<<(N * DIM + 255) / 256, blk, 0, stream>>>(a_s, a_d, m, den, h, gat_bias, gat, N);
  // 4) Laplacian norm via two SCALAR SpMMs (128x less edge traffic)
  row_sum    <<<waveBlocks, blk, 0, stream>>>(gat, rsum, N);
  spmm_scalar<<<2048, blk, 0, stream>>>(src, dst, deg, rsum, macc, E);
  row_q      <<<waveBlocks, blk, 0, stream>>>(gat, macc, meanv, qv, N);
  spmm_scalar<<<2048, blk, 0, stream>>>(src, dst, deg, qv, vacc, E);
  // 5) lap-norm affine + residual + LayerNorm (fused)
  lap_ln<<<waveBlocks, blk, 0, stream>>>(gat, meanv, vacc, ln1s, ln1b, ln2s, ln2b, x, out2, N);
  // 6) MLP on WMMA: hidden = gelu(out2@W1 + b1); out = hidden@W2 + b2 + out2
  wmma_gemm<DIM, 1>    <<<dim3(rowTiles, 2), blk, 0, stream>>>(out2, W1, b1, nullptr, hidden, 2 * DIM);
  wmma_gemm<2 * DIM, 2><<<dim3(rowTiles, 1), blk, 0, stream>>>(hidden, W2, b2, out2, (float*)d_out, DIM);
  // 7) row L2 normalize
  l2norm<<<waveBlocks, blk, 0, stream>>>((float*)d_out, N);
}